// DynamicReceptiveEncoder_34969623724329
// MI455X (gfx1250) — compile-verified
//
#include <hip/hip_runtime.h>
#include <hip/hip_bf16.h>

typedef __attribute__((ext_vector_type(2))) float v2f;
typedef __attribute__((ext_vector_type(8))) float v8f;

#define C_RAW 51
#define CCH   64
#define TLEN  2048
#define BB    64
#define KTOT  153          // 51 * 3
#define KP    156          // padded to multiple of 4 (39 wmma k-steps)
#define TTILE 128          // t columns per block
#define XSTR  133          // padded LDS row stride for X tile (bank-friendly)

// ---------------------------------------------------------------------------
// Kernel 1: conv1d (kernel 3, pad 1) + bias as an fp32 WMMA GEMM.
// act[b,c,t] = sum_{i,k} inputs[b,t+k-1,i] * w[c,i,k] + bias[c]
// A = W tile (16x4), B = X tile (4x16), D = 16x16 fp32, 39 k-steps.
// ---------------------------------------------------------------------------
__global__ __launch_bounds__(256) void conv1d_wmma_kernel(
    const float* __restrict__ inp,   // [B, T, 51]
    const float* __restrict__ w1d,   // [64, 51, 3]
    const float* __restrict__ b1d,   // [64]
    float* __restrict__ act) {       // [B, 64, T]
  extern __shared__ float smem[];
  float* Ws = smem;                  // [64][156]
  float* Xs = smem + CCH * KP;       // [156][XSTR]
  const int tid = threadIdx.x;
  const int b   = blockIdx.x >> 4;
  const int t0  = (blockIdx.x & 15) * TTILE;

  // Stage weights: Ws[c][kk], kk = k*51 + i, zero-padded to 156.
  for (int idx = tid; idx < CCH * KP; idx += 256) {
    int c  = idx / KP;
    int kk = idx - c * KP;
    float v = 0.f;
    if (kk < KTOT) {
      int k = kk / C_RAW;
      int i = kk - k * C_RAW;
      v = w1d[c * KTOT + i * 3 + k];
    }
    Ws[c * KP + kk] = v;
  }
  // Stage X tile: Xs[kk][jj] = inputs[b, t0+jj+k-1, i] (zero outside [0,T)).
  for (int idx = tid; idx < KP * TTILE; idx += 256) {
    int jj = idx / KP;
    int kk = idx - jj * KP;
    float v = 0.f;
    if (kk < KTOT) {
      int k = kk / C_RAW;
      int i = kk - k * C_RAW;
      int t = t0 + jj + k - 1;
      if ((unsigned)t < (unsigned)TLEN) v = inp[(b * TLEN + t) * C_RAW + i];
    }
    Xs[kk * XSTR + jj] = v;
  }
  __syncthreads();

  const int wv    = tid >> 5;          // wave id 0..7 -> N tile
  const int ln    = tid & 31;
  const int nrow  = ln & 15;           // M row (A) / N col (B,D)
  const int khalf = (ln >> 4) << 1;    // lanes 16-31 hold K = 2,3
  const int n0    = wv * 16;

  v8f acc[4] = {v8f{}, v8f{}, v8f{}, v8f{}};
  for (int k0 = 0; k0 < KP; k0 += 4) {
    const int kr = k0 + khalf;
    v2f bf;
    bf.x = Xs[kr * XSTR + n0 + nrow];
    bf.y = Xs[(kr + 1) * XSTR + n0 + nrow];
#pragma unroll
    for (int m = 0; m < 4; ++m) {
      v2f af = *(const v2f*)&Ws[(m * 16 + nrow) * KP + kr];
      acc[m] = __builtin_amdgcn_wmma_f32_16x16x4_f32(
          false, af, false, bf, (short)0, acc[m], false, false);
    }
  }

  const int half = (ln >> 4) * 8;      // lanes 16-31 hold rows M+8
  const int tcol = t0 + n0 + nrow;
#pragma unroll
  for (int m = 0; m < 4; ++m) {
#pragma unroll
    for (int r = 0; r < 8; ++r) {
      int c = m * 16 + half + r;
      act[(b * CCH + c) * TLEN + tcol] = acc[m][r] + b1d[c];
    }
  }
}

// ---------------------------------------------------------------------------
// Kernel 2: per-batch LayerNorm statistics over (C,T) = 131072 elements.
// ---------------------------------------------------------------------------
__global__ __launch_bounds__(256) void ln_stats_kernel(
    const float* __restrict__ act, float* __restrict__ st) {
  __shared__ float r1[256];
  __shared__ float r2[256];
  const int b = blockIdx.x, tid = threadIdx.x;
  const float* p = act + (size_t)b * CCH * TLEN;
  float s = 0.f, s2 = 0.f;
  for (int i = tid; i < CCH * TLEN; i += 256) {
    float v = p[i];
    s += v;
    s2 = fmaf(v, v, s2);
  }
  r1[tid] = s; r2[tid] = s2;
  __syncthreads();
  for (int off = 128; off > 0; off >>= 1) {
    if (tid < off) { r1[tid] += r1[tid + off]; r2[tid] += r2[tid + off]; }
    __syncthreads();
  }
  if (tid == 0) {
    const float inv = 1.f / (float)(CCH * TLEN);
    float mu  = r1[0] * inv;
    float var = r2[0] * inv - mu * mu;
    st[2 * b]     = mu;
    st[2 * b + 1] = rsqrtf(var + 1e-5f);
  }
}

// ---------------------------------------------------------------------------
// Kernel 3: fused LN-apply + delta + 3x SlidingPSN + heaviside + 1x1 conv.
// Flat-reshape identities: at scan step u, lane L (0..4095):
//   flat = u*4096 + L ; b = u>>5 (uniform) ; c = (2u + (L>>11)) & 63 ;
//   t = L & 2047 (constant per lane) ; output -> out[L*2048 + u].
// ---------------------------------------------------------------------------
__global__ __launch_bounds__(128) void psn_scan_kernel(
    const float* __restrict__ act, const float* __restrict__ st,
    const float* __restrict__ lnw, const float* __restrict__ lnb,
    const float* __restrict__ wsp, const float* __restrict__ bsp,
    const float* __restrict__ wfp, const float* __restrict__ bfp,
    const float* __restrict__ wnp, const float* __restrict__ bnp,
    const float* __restrict__ c2w, const float* __restrict__ c2b,
    float* __restrict__ out) {
  const int lane = blockIdx.x * 128 + threadIdx.x;   // 0..4095
  const int t    = lane & 2047;
  const int hi   = lane >> 11;
  const bool tv  = (t != 0);

  float wsv[16], wfv[8], wnv[4];
#pragma unroll
  for (int i = 0; i < 16; ++i) wsv[i] = wsp[i];
#pragma unroll
  for (int i = 0; i < 8; ++i) wfv[i] = wfp[i];
#pragma unroll
  for (int i = 0; i < 4; ++i) wnv[i] = wnp[i];
  const float bsc = bsp[0], bfc = bfp[0], bnc = bnp[0];
  const float w0 = c2w[0], w1 = c2w[1], w2 = c2w[2], cbv = c2b[0];

  float qs[16] = {}, qf[8] = {}, qn[4] = {};
  float prev_s = 0.f, prev_n = 0.f;

  for (int u0 = 0; u0 < TLEN; u0 += 8) {
    const int bidx = u0 >> 5;                 // uniform within the 8-group
    const float mu = st[2 * bidx];
    const float rs = st[2 * bidx + 1];

    float a0[8], a1[8], lw0v[8], lb0v[8], lw1v[8], lb1v[8];
#pragma unroll
    for (int j = 0; j < 8; ++j) {
      const int u    = u0 + j;
      const int flat = u * 4096 + lane;
      a0[j] = act[flat];
      const int c  = (2 * u + hi) & 63;
      const int li = c * 2048 + t;
      lw0v[j] = lnw[li];
      lb0v[j] = lnb[li];
      if (tv) {
        a1[j]   = act[flat - 1];
        lw1v[j] = lnw[li - 1];
        lb1v[j] = lnb[li - 1];
      } else {
        a1[j] = 0.f; lw1v[j] = 0.f; lb1v[j] = 0.f;
      }
    }
    if (u0 + 8 < TLEN)
      __builtin_prefetch(act + (u0 + 8) * 4096 + lane, 0, 1);

    float og[8];
#pragma unroll
    for (int j = 0; j < 8; ++j) {
      const float an = (a0[j] - mu) * rs * lw0v[j] + lb0v[j];
      float d = 0.f;
      if (tv) d = an - ((a1[j] - mu) * rs * lw1v[j] + lb1v[j]);

      // shift registers (fully unrolled -> register renames)
#pragma unroll
      for (int z = 0; z < 15; ++z) qs[z] = qs[z + 1];
#pragma unroll
      for (int z = 0; z < 7; ++z) qf[z] = qf[z + 1];
#pragma unroll
      for (int z = 0; z < 3; ++z) qn[z] = qn[z + 1];

      // partial dots over the 15/7/3 history taps (off the feedback path)
      float ps = bsc;
#pragma unroll
      for (int z = 0; z < 15; ++z) ps = fmaf(wsv[z], qs[z], ps);
      float pf = bfc;
#pragma unroll
      for (int z = 0; z < 7; ++z) pf = fmaf(wfv[z], qf[z], pf);
      float pn = bnc;
#pragma unroll
      for (int z = 0; z < 3; ++z) pn = fmaf(wnv[z], qn[z], pn);

      // feedback terms: weight of newest tap is 1.0 (exp_init)
      const float s_in = fmaf(-0.5f, prev_s, an);      // ADAPT_RATE
      qs[15] = s_in;
      const float s = (fmaf(wsv[15], s_in, ps) >= 0.f) ? 1.f : 0.f;

      qf[7] = d;
      const float f = (fmaf(wfv[7], d, pf) >= 0.f) ? 1.f : 0.f;

      const float n_in = fmaf(0.5f, prev_n, an);       // SENSITIZE_RATE
      qn[3] = n_in;
      const float n = (fmaf(wnv[3], n_in, pn) >= 0.f) ? 1.f : 0.f;

      prev_s = s;
      prev_n = n;
      og[j] = fmaf(w0, s, fmaf(w1, f, fmaf(w2, n, cbv)));
    }

    float4* op = (float4*)(out + (size_t)lane * TLEN + u0);
    op[0] = make_float4(og[0], og[1], og[2], og[3]);
    op[1] = make_float4(og[4], og[5], og[6], og[7]);
  }
}

// ---------------------------------------------------------------------------
extern "C" void kernel_launch(void* const* d_in, const int* in_sizes, int n_in,
                              void* d_out, int out_size, void* d_ws, size_t ws_size,
                              hipStream_t stream) {
  const float* inp = (const float*)d_in[0];
  const float* w1d = (const float*)d_in[1];
  const float* b1d = (const float*)d_in[2];
  const float* lnw = (const float*)d_in[3];
  const float* lnb = (const float*)d_in[4];
  const float* wsp = (const float*)d_in[5];
  const float* bsp = (const float*)d_in[6];
  const float* wfp = (const float*)d_in[7];
  const float* bfp = (const float*)d_in[8];
  const float* wnp = (const float*)d_in[9];
  const float* bnp = (const float*)d_in[10];
  const float* c2w = (const float*)d_in[11];
  const float* c2b = (const float*)d_in[12];

  float* out = (float*)d_out;
  float* act = (float*)d_ws;                       // 64*64*2048 floats
  float* st  = act + (size_t)BB * CCH * TLEN;      // 64 * {mu, rstd}

  const size_t smem = (size_t)(CCH * KP + KP * XSTR) * sizeof(float);
  conv1d_wmma_kernel<<<dim3(BB * (TLEN / TTILE)), dim3(256), smem, stream>>>(
      inp, w1d, b1d, act);
  ln_stats_kernel<<<dim3(BB), dim3(256), 0, stream>>>(act, st);
  psn_scan_kernel<<<dim3(32), dim3(128), 0, stream>>>(
      act, st, lnw, lnb, wsp, bsp, wfp, bfp, wnp, bnp, c2w, c2b, out);
}